// ConstraintLayer_14456859918871
// MI455X (gfx1250) — compile-verified
//
#include <hip/hip_runtime.h>
#include <hip/hip_bf16.h>
#include <math.h>

// Problem constants (match reference)
constexpr int B_  = 4096;
constexpr int IN_ = 2048;
constexpr int N_  = 512;
constexpr int K_  = 1024;
constexpr int NQ_ = 8;
constexpr int MR_ = 1024;   // 2*N rows of D

typedef float v2f __attribute__((ext_vector_type(2)));
typedef float v8f __attribute__((ext_vector_type(8)));

// ---------------------------------------------------------------------------
// WMMA f32 NT GEMM:  C[m,n] = sum_k A[m,k] * Bt[n,k]  (+ bias[n] if given)
// Block: 256 threads (8 waves). Block tile 128(M) x 128(N), K-stage 16.
// Wave grid 4(M) x 2(N); each wave computes 32x64 = 2x4 WMMA 16x16 tiles.
// Double-buffered LDS; global loads for stage s+1 overlap WMMAs of stage s.
// ---------------------------------------------------------------------------
#define TILE_M 128
#define TILE_N 128
#define TILE_K 16
#define LDS_STRIDE 20   // floats; 80B row stride: 16B-aligned rows, conflict-free frags

__global__ __launch_bounds__(256)
void wmma_gemm_nt(const float* __restrict__ A, const float* __restrict__ Bt,
                  const float* __restrict__ bias, float* __restrict__ C,
                  int M, int Nn, int Kk, int lda, int ldb, int ldc)
{
    __shared__ __align__(16) float As[2][TILE_M * LDS_STRIDE];
    __shared__ __align__(16) float Bs[2][TILE_N * LDS_STRIDE];

    const int tid   = threadIdx.x;
    const int lane  = tid & 31;
    const int wave  = tid >> 5;     // 0..7
    const int wm    = wave >> 1;    // 0..3 (M direction, 32 rows each)
    const int wn    = wave & 1;     // 0..1 (N direction, 64 cols each)
    const int lhalf = lane >> 4;    // 0/1
    const int lmod  = lane & 15;

    const int m0 = blockIdx.y * TILE_M;
    const int n0 = blockIdx.x * TILE_N;

    // global->LDS staging assignment: 512 float4 per 128x16 tile, 256 threads
    const int row0 = tid >> 2;            // 0..63
    const int cg   = (tid & 3) << 2;      // 0,4,8,12
    const int row1 = row0 + 64;           // 64..127

    const float* Ap0 = A  + (size_t)(m0 + row0) * lda + cg;
    const float* Ap1 = A  + (size_t)(m0 + row1) * lda + cg;
    const float* Bp0 = Bt + (size_t)(n0 + row0) * ldb + cg;
    const float* Bp1 = Bt + (size_t)(n0 + row1) * ldb + cg;

    const int nstages = Kk / TILE_K;

    float4 ra0, ra1, rb0, rb1;

    // ---- prologue: stage 0 ----
    ra0 = *(const float4*)(Ap0);
    ra1 = *(const float4*)(Ap1);
    rb0 = *(const float4*)(Bp0);
    rb1 = *(const float4*)(Bp1);
    *(float4*)&As[0][row0 * LDS_STRIDE + cg] = ra0;
    *(float4*)&As[0][row1 * LDS_STRIDE + cg] = ra1;
    *(float4*)&Bs[0][row0 * LDS_STRIDE + cg] = rb0;
    *(float4*)&Bs[0][row1 * LDS_STRIDE + cg] = rb1;
    __syncthreads();

    v8f acc[2][4] = {};

    for (int s = 0; s < nstages; ++s) {
        const int cur = s & 1;
        const bool hasNext = (s + 1 < nstages);

        // issue global loads for next stage (overlap with WMMAs below)
        if (hasNext) {
            const int ko = (s + 1) * TILE_K;
            ra0 = *(const float4*)(Ap0 + ko);
            ra1 = *(const float4*)(Ap1 + ko);
            rb0 = *(const float4*)(Bp0 + ko);
            rb1 = *(const float4*)(Bp1 + ko);
        }
        // prefetch two stages ahead (global_prefetch_b8 on gfx1250)
        if (s + 2 < nstages) {
            const int kp = (s + 2) * TILE_K;
            __builtin_prefetch(Ap0 + kp, 0, 1);
            __builtin_prefetch(Bp0 + kp, 0, 1);
        }

        // ---- compute: 4 k-steps of 4, 32 WMMAs per wave per stage ----
#pragma unroll
        for (int kk = 0; kk < TILE_K; kk += 4) {
            v2f af[2], bf[4];
#pragma unroll
            for (int tm = 0; tm < 2; ++tm) {
                int row = wm * 32 + tm * 16 + lmod;
                af[tm] = *(const v2f*)&As[cur][row * LDS_STRIDE + kk + 2 * lhalf];
            }
#pragma unroll
            for (int tn = 0; tn < 4; ++tn) {
                int row = wn * 64 + tn * 16 + lmod;
                bf[tn] = *(const v2f*)&Bs[cur][row * LDS_STRIDE + kk + 2 * lhalf];
            }
#pragma unroll
            for (int tm = 0; tm < 2; ++tm)
#pragma unroll
                for (int tn = 0; tn < 4; ++tn)
                    acc[tm][tn] = __builtin_amdgcn_wmma_f32_16x16x4_f32(
                        false, af[tm], false, bf[tn],
                        (short)0, acc[tm][tn], false, false);
        }

        // ---- stage registers into the other LDS buffer ----
        if (hasNext) {
            const int nxt = cur ^ 1;
            *(float4*)&As[nxt][row0 * LDS_STRIDE + cg] = ra0;
            *(float4*)&As[nxt][row1 * LDS_STRIDE + cg] = ra1;
            *(float4*)&Bs[nxt][row0 * LDS_STRIDE + cg] = rb0;
            *(float4*)&Bs[nxt][row1 * LDS_STRIDE + cg] = rb1;
        }
        __syncthreads();
    }

    // ---- epilogue: C layout vgpr v -> M = v + 8*lhalf, N = lmod ----
#pragma unroll
    for (int tm = 0; tm < 2; ++tm) {
#pragma unroll
        for (int tn = 0; tn < 4; ++tn) {
            int n = n0 + wn * 64 + tn * 16 + lmod;
            float bv = bias ? bias[n] : 0.0f;
#pragma unroll
            for (int v = 0; v < 8; ++v) {
                int m = m0 + wm * 32 + tm * 16 + v + 8 * lhalf;
                C[(size_t)m * ldc + n] = acc[tm][tn][v] + bv;
            }
        }
    }
}

// ---------------------------------------------------------------------------
// Small helper kernels
// ---------------------------------------------------------------------------

// w[k] = dot(NA_E[k,:], z0) + y1[k]
__global__ __launch_bounds__(256)
void w_kernel(const float* __restrict__ NA_E, const float* __restrict__ z0,
              const float* __restrict__ y1, float* __restrict__ w)
{
    int k = blockIdx.x * 256 + threadIdx.x;
    if (k >= K_) return;
    float s = 0.f;
    const float* row = NA_E + (size_t)k * N_;
    for (int j = 0; j < N_; ++j) s += row[j] * z0[j];
    w[k] = s + y1[k];
}

// Pw[i,k] = dot(P[i][k][:], w);  Pwq = Pw + q
__global__ __launch_bounds__(256)
void pw_kernel(const float* __restrict__ P, const float* __restrict__ w,
               const float* __restrict__ q, float* __restrict__ Pw,
               float* __restrict__ Pwq)
{
    int idx = blockIdx.x * 256 + threadIdx.x;   // 0 .. NQ*K-1
    if (idx >= NQ_ * K_) return;
    const float* row = P + (size_t)idx * K_;
    float s = 0.f;
    for (int j = 0; j < K_; ++j) s += row[j] * w[j];
    Pw[idx]  = s;
    Pwq[idx] = s + q[idx];
}

// c[i] = sum_k w[k]*(0.5*Pw[i,k] + q[i,k]) + r[i]
__global__ __launch_bounds__(256)
void c_kernel(const float* __restrict__ w, const float* __restrict__ Pw,
              const float* __restrict__ q, const float* __restrict__ r,
              float* __restrict__ c)
{
    __shared__ float red[256];
    int i = blockIdx.x;
    float s = 0.f;
    for (int k = threadIdx.x; k < K_; k += 256)
        s += w[k] * (0.5f * Pw[i * K_ + k] + q[i * K_ + k]);
    red[threadIdx.x] = s; __syncthreads();
    for (int off = 128; off > 0; off >>= 1) {
        if (threadIdx.x < off) red[threadIdx.x] += red[threadIdx.x + off];
        __syncthreads();
    }
    if (threadIdx.x == 0) c[i] = red[0] + r[i];
}

// beta[b] = dot(x[b,:], wrow) + bias0
__global__ __launch_bounds__(256)
void beta_kernel(const float* __restrict__ x, const float* __restrict__ wrow,
                 const float* __restrict__ bias0, float* __restrict__ beta)
{
    __shared__ float red[256];
    int b = blockIdx.x;
    const float* xr = x + (size_t)b * IN_;
    float s = 0.f;
    for (int k = threadIdx.x; k < IN_; k += 256) s += xr[k] * wrow[k];
    red[threadIdx.x] = s; __syncthreads();
    for (int off = 128; off > 0; off >>= 1) {
        if (threadIdx.x < off) red[threadIdx.x] += red[threadIdx.x + off];
        __syncthreads();
    }
    if (threadIdx.x == 0) beta[b] = red[0] + bias0[0];
}

// in-place row normalize: v_bar = v / max(||v||, 1e-12)   (row length N_=512)
__global__ __launch_bounds__(256)
void normalize_kernel(float* __restrict__ v)
{
    __shared__ float red[256];
    int b = blockIdx.x;
    float* row = v + (size_t)b * N_;
    float vals[2];
    float s = 0.f;
#pragma unroll
    for (int i = 0; i < 2; ++i) {
        vals[i] = row[threadIdx.x + i * 256];
        s += vals[i] * vals[i];
    }
    red[threadIdx.x] = s; __syncthreads();
    for (int off = 128; off > 0; off >>= 1) {
        if (threadIdx.x < off) red[threadIdx.x] += red[threadIdx.x + off];
        __syncthreads();
    }
    float inv = 1.0f / fmaxf(sqrtf(red[0]), 1e-12f);
#pragma unroll
    for (int i = 0; i < 2; ++i) row[threadIdx.x + i * 256] = vals[i] * inv;
}

// kappa[b] = relu(max_k T[b,k])   (row length Kc)
__global__ __launch_bounds__(256)
void rowmax_relu_kernel(const float* __restrict__ T, float* __restrict__ kappa, int Kc)
{
    __shared__ float red[256];
    int b = blockIdx.x;
    float m = -3.4e38f;
    const float* row = T + (size_t)b * Kc;
    for (int k = threadIdx.x; k < Kc; k += 256) m = fmaxf(m, row[k]);
    red[threadIdx.x] = m; __syncthreads();
    for (int off = 128; off > 0; off >>= 1) {
        if (threadIdx.x < off) red[threadIdx.x] = fmaxf(red[threadIdx.x], red[threadIdx.x + off]);
        __syncthreads();
    }
    if (threadIdx.x == 0) kappa[b] = fmaxf(red[0], 0.0f);
}

// a[b,i] = 0.5*dot(S[b,:], rho[b,:]);  bq[b,i] = dot(rho[b,:], Pwq_i[:])
__global__ __launch_bounds__(256)
void abq_kernel(const float* __restrict__ S, const float* __restrict__ rho,
                const float* __restrict__ Pwq_i, float* __restrict__ a,
                float* __restrict__ bq, int i)
{
    __shared__ float ra[256];
    __shared__ float rb[256];
    int b = blockIdx.x;
    const float* Sr = S   + (size_t)b * K_;
    const float* Rr = rho + (size_t)b * K_;
    float sa = 0.f, sb = 0.f;
    for (int k = threadIdx.x; k < K_; k += 256) {
        float rv = Rr[k];
        sa += Sr[k] * rv;
        sb += rv * Pwq_i[k];
    }
    ra[threadIdx.x] = sa; rb[threadIdx.x] = sb; __syncthreads();
    for (int off = 128; off > 0; off >>= 1) {
        if (threadIdx.x < off) {
            ra[threadIdx.x] += ra[threadIdx.x + off];
            rb[threadIdx.x] += rb[threadIdx.x + off];
        }
        __syncthreads();
    }
    if (threadIdx.x == 0) {
        a [b * NQ_ + i] = 0.5f * ra[0];
        bq[b * NQ_ + i] = rb[0];
    }
}

// lam roots -> kappa update -> alpha[b] = 1/(exp(beta)+kappa)
__global__ __launch_bounds__(256)
void alpha_kernel(const float* __restrict__ beta, const float* __restrict__ kappa,
                  const float* __restrict__ a, const float* __restrict__ bq,
                  const float* __restrict__ c, float* __restrict__ alpha)
{
    int b = blockIdx.x * 256 + threadIdx.x;
    if (b >= B_) return;
    float kap = kappa[b];
    float mx = -3.4e38f;
#pragma unroll
    for (int i = 0; i < NQ_; ++i) {
        float ai = a[b * NQ_ + i];
        float bi = bq[b * NQ_ + i];
        float ci = c[i];
        float disc = bi * bi - 4.0f * ai * ci;
        float lam = (-bi + sqrtf(disc)) / (2.0f * ai);
        mx = fmaxf(mx, 1.0f / lam);
    }
    kap = fmaxf(kap, mx);
    alpha[b] = 1.0f / (expf(beta[b]) + kap);
}

// y[b,k] = alpha[b]*rho[b,k] + w[k]
__global__ __launch_bounds__(256)
void y_kernel(const float* __restrict__ rho, const float* __restrict__ alpha,
              const float* __restrict__ w, float* __restrict__ y)
{
    size_t idx = (size_t)blockIdx.x * 256 + threadIdx.x;
    int b = (int)(idx >> 10);        // /K_
    int k = (int)(idx & (K_ - 1));
    y[idx] = alpha[b] * rho[idx] + w[k];
}

// ---------------------------------------------------------------------------
extern "C" void kernel_launch(void* const* d_in, const int* in_sizes, int n_in,
                              void* d_out, int out_size, void* d_ws, size_t ws_size,
                              hipStream_t stream)
{
    const float* x     = (const float*)d_in[0];
    const float* W_map = (const float*)d_in[1];
    const float* b_map = (const float*)d_in[2];
    const float* Dm    = (const float*)d_in[3];
    const float* NA_E  = (const float*)d_in[4];
    const float* y1    = (const float*)d_in[5];
    const float* z0    = (const float*)d_in[6];
    const float* all_P = (const float*)d_in[7];
    const float* all_q = (const float*)d_in[8];
    const float* all_r = (const float*)d_in[9];
    float* y_out = (float*)d_out;

    float* ws    = (float*)d_ws;
    float* vbar  = ws;                          // B*N
    float* rho   = vbar + (size_t)B_ * N_;      // B*K
    float* S     = rho  + (size_t)B_ * K_;      // B*K (reused scratch)
    float* beta  = S    + (size_t)B_ * K_;      // B
    float* kappa = beta  + B_;                  // B
    float* alpha = kappa + B_;                  // B
    float* a     = alpha + B_;                  // B*NQ
    float* bq    = a     + B_ * NQ_;            // B*NQ
    float* w     = bq    + B_ * NQ_;            // K
    float* Pw    = w     + K_;                  // NQ*K
    float* Pwq   = Pw    + NQ_ * K_;            // NQ*K
    float* cbuf  = Pwq   + NQ_ * K_;            // NQ

    // --- small precompute ---
    w_kernel <<<K_ / 256, 256, 0, stream>>>(NA_E, z0, y1, w);
    pw_kernel<<<(NQ_ * K_) / 256, 256, 0, stream>>>(all_P, w, all_q, Pw, Pwq);
    c_kernel <<<NQ_, 256, 0, stream>>>(w, Pw, all_q, all_r, cbuf);

    // --- beta = x @ W_map[N].T + b_map[N] ---
    beta_kernel<<<B_, 256, 0, stream>>>(x, W_map + (size_t)N_ * IN_, b_map + N_, beta);

    // --- GEMM1: v = x @ W_map[:N].T + b_map[:N]  (into vbar, then normalize) ---
    {
        dim3 grid(N_ / TILE_N, B_ / TILE_M);
        wmma_gemm_nt<<<grid, 256, 0, stream>>>(x, W_map, b_map, vbar,
                                               B_, N_, IN_, IN_, IN_, N_);
    }
    normalize_kernel<<<B_, 256, 0, stream>>>(vbar);

    // --- GEMM2: T = vbar @ D.T -> S; kappa = relu(rowmax) ---
    {
        dim3 grid(MR_ / TILE_N, B_ / TILE_M);
        wmma_gemm_nt<<<grid, 256, 0, stream>>>(vbar, Dm, nullptr, S,
                                               B_, MR_, N_, N_, N_, MR_);
    }
    rowmax_relu_kernel<<<B_, 256, 0, stream>>>(S, kappa, MR_);

    // --- GEMM3: rho = vbar @ NA_E.T ---
    {
        dim3 grid(K_ / TILE_N, B_ / TILE_M);
        wmma_gemm_nt<<<grid, 256, 0, stream>>>(vbar, NA_E, nullptr, rho,
                                               B_, K_, N_, N_, N_, K_);
    }

    // --- per-i quadratic forms: S = rho @ P_i (P symmetric); a, bq ---
    for (int i = 0; i < NQ_; ++i) {
        dim3 grid(K_ / TILE_N, B_ / TILE_M);
        wmma_gemm_nt<<<grid, 256, 0, stream>>>(rho, all_P + (size_t)i * K_ * K_,
                                               nullptr, S, B_, K_, K_, K_, K_, K_);
        abq_kernel<<<B_, 256, 0, stream>>>(S, rho, Pwq + (size_t)i * K_, a, bq, i);
    }

    // --- alpha, then y = alpha*rho + w ---
    alpha_kernel<<<(B_ + 255) / 256, 256, 0, stream>>>(beta, kappa, a, bq, cbuf, alpha);
    y_kernel<<<(B_ * K_) / 256, 256, 0, stream>>>(rho, alpha, w, y_out);
}